// BinaryLoss_6502580486642
// MI455X (gfx1250) — compile-verified
//
#include <hip/hip_runtime.h>
#include <hip/hip_bf16.h>
#include <stdint.h>

// Problem constants (from reference: N_PTS=16384, N=16384, K=17)
#define NROWS 16384
#define NPTS  16384
#define MPAIR 256            // (K-1)^2
#define NTRI  32768          // 2*N triangles
#define PBIG  16385ULL       // num_pts + 1
#define PBIG2 (PBIG*PBIG)

// Workspace layout (bytes). Total ~34.3 MB.
#define OFF_KEYS   0u                    // 32768 * 8 B  = 262144
#define OFF_COUNTS 262144u               // 16400 * 4 B  = 65600
#define OFF_BITS   327744u               // 16384*16384 bits = 33554432 B
#define OFF_ROWS   33882176u             // 5 * 16384 * 4 B
#define ZERO_WORDS (16400u + 8388608u)   // counts + bitmask words (contiguous)

typedef float v2f __attribute__((ext_vector_type(2)));
typedef float v8f __attribute__((ext_vector_type(8)));

__device__ __forceinline__ float softplusf(float x) {
    // numerically stable log(1+exp(x))
    return fmaxf(x, 0.f) + log1pf(expf(-fabsf(x)));
}

__device__ __forceinline__ void sort3i(int& x, int& y, int& z) {
    int t;
    if (x > y) { t = x; x = y; y = t; }
    if (y > z) { t = y; y = z; z = t; }
    if (x > y) { t = x; x = y; y = t; }
}

__device__ __forceinline__ int edgebit(const unsigned* __restrict__ bits,
                                       unsigned u, unsigned v) {
    unsigned idx = (u << 14) | v;  // u*16384 + v  (< 2^28)
    return (int)((bits[idx >> 5] >> (idx & 31u)) & 1u);
}

__device__ __forceinline__ bool key_member(const unsigned long long* __restrict__ ks,
                                           unsigned long long key) {
    int lo = 0, hi = NTRI;          // lower_bound over sorted keys (dups fine)
    while (lo < hi) {
        int mid = (lo + hi) >> 1;
        if (ks[mid] < key) lo = mid + 1; else hi = mid;
    }
    return (lo < NTRI) && (ks[lo] == key);
}

// ---------------- kernels ----------------

__global__ void zero_words(unsigned* __restrict__ w, unsigned n) {
    unsigned i = blockIdx.x * blockDim.x + threadIdx.x;
    if (i < n) w[i] = 0u;
}

// sorted-vertex key per triangle
__global__ void tri_keys(const long long* __restrict__ tri,
                         unsigned long long* __restrict__ keys) {
    int i = blockIdx.x * blockDim.x + threadIdx.x;
    if (i >= NTRI) return;
    int a = (int)tri[3 * i + 0];
    int b = (int)tri[3 * i + 1];
    int c = (int)tri[3 * i + 2];
    sort3i(a, b, c);
    keys[i] = (unsigned long long)a * PBIG2 +
              (unsigned long long)b * PBIG +
              (unsigned long long)c;
}

// set all 6 ordered-pair edge bits per triangle (idempotent -> deterministic)
__global__ void edge_bits(const long long* __restrict__ tri,
                          unsigned* __restrict__ bits) {
    int i = blockIdx.x * blockDim.x + threadIdx.x;
    if (i >= NTRI) return;
    unsigned a = (unsigned)tri[3 * i + 0];
    unsigned b = (unsigned)tri[3 * i + 1];
    unsigned c = (unsigned)tri[3 * i + 2];
    unsigned e[6];
    e[0] = (a << 14) | b; e[1] = (b << 14) | a;
    e[2] = (b << 14) | c; e[3] = (c << 14) | b;
    e[4] = (c << 14) | a; e[5] = (a << 14) | c;
    #pragma unroll
    for (int k = 0; k < 6; ++k)
        atomicOr(&bits[e[k] >> 5], 1u << (e[k] & 31u));
}

// one compare-exchange pass of a global bitonic sort over 32768 u64 keys
__global__ void bitonic_step(unsigned long long* __restrict__ keys, int j, int k) {
    int i = blockIdx.x * blockDim.x + threadIdx.x;
    if (i >= NTRI) return;
    int l = i ^ j;
    if (l > i) {
        unsigned long long x = keys[i], y = keys[l];
        bool up = ((i & k) == 0);
        if ((x > y) == up) { keys[i] = y; keys[l] = x; }
    }
}

// per-vertex unique-triangle counts (dedup via first-occurrence in sorted keys)
__global__ void count_unique(const unsigned long long* __restrict__ keys,
                             unsigned* __restrict__ counts) {
    int i = blockIdx.x * blockDim.x + threadIdx.x;
    if (i >= NTRI) return;
    unsigned long long k0 = keys[i];
    if (i > 0 && keys[i - 1] == k0) return;   // duplicate triangle
    unsigned v0 = (unsigned)(k0 / PBIG2);
    unsigned long long r = k0 % PBIG2;
    unsigned v1 = (unsigned)(r / PBIG);
    unsigned v2 = (unsigned)(r % PBIG);
    atomicAdd(&counts[v0], 1u);
    if (v1 != v0) atomicAdd(&counts[v1], 1u);
    if (v2 != v1) atomicAdd(&counts[v2], 1u);
}

// one wave (32 lanes) per row; 8 waves per block
__global__ __launch_bounds__(256)
void main_rows(const float* __restrict__ pred,
               const long long* __restrict__ knn,
               const unsigned long long* __restrict__ keys,
               const unsigned* __restrict__ counts,
               const unsigned* __restrict__ bits,
               float* __restrict__ rows) {
    const int wave = threadIdx.x >> 5;
    const int lane = threadIdx.x & 31;
    const int row  = blockIdx.x * 8 + wave;

    const long long* kr = knn + (long long)row * 17;
    const int c = (int)kr[0];
    int a[16];
    #pragma unroll
    for (int j = 0; j < 16; ++j) a[j] = (int)kr[1 + j];

    const float* pr = pred + (long long)row * MPAIR;

    // ---- phase 1: BCE terms + masked-label count over 256 pairs ----
    float sp_sum = 0.f, xg_sum = 0.f;
    int cnt = 0;
    #pragma unroll
    for (int t = 0; t < 8; ++t) {
        const int p  = lane + 32 * t;       // coalesced over pred row
        const int e0 = a[p >> 4];
        const int e1 = a[p & 15];
        int g = edgebit(bits, (unsigned)e0, (unsigned)e1);
        if (g) {                            // rare: only then binary-search
            int v0 = c, v1 = e0, v2 = e1;
            sort3i(v0, v1, v2);
            unsigned long long key = (unsigned long long)v0 * PBIG2 +
                                     (unsigned long long)v1 * PBIG +
                                     (unsigned long long)v2;
            g = key_member(keys, key) ? 1 : 0;
        }
        const float x = pr[p];
        sp_sum += softplusf(x);
        if (g) { xg_sum += x; cnt += 1; }
    }
    #pragma unroll
    for (int o = 16; o >= 1; o >>= 1) {
        sp_sum += __shfl_xor(sp_sum, o);
        xg_sum += __shfl_xor(xg_sum, o);
        cnt    += __shfl_xor(cnt, o);
    }

    // ---- phase 2: top-3 of each 16-wide sub-row (lanes 0..15) ----
    float pp = 0.f, nn = 0.f;
    int gc = 0;
    if (lane < 16) {
        const float* q = pr + lane * 16;    // row is L0/L2-hot from phase 1
        float m1 = -INFINITY, m2 = -INFINITY, m3 = -INFINITY;
        #pragma unroll
        for (int i = 0; i < 16; ++i) {
            float v = q[i];
            if (v > m1)      { m3 = m2; m2 = m1; m1 = v; }
            else if (v > m2) { m3 = m2; m2 = v; }
            else if (v > m3) { m3 = v; }
        }
        if (edgebit(bits, (unsigned)c, (unsigned)a[lane])) {
            gc = 1;
            pp = softplusf(-m2);            // sp(-pos_pred)
            nn = softplusf(m3);             // sp(neg_pred)
        }
    }
    #pragma unroll
    for (int o = 16; o >= 1; o >>= 1) {
        pp += __shfl_xor(pp, o);
        nn += __shfl_xor(nn, o);
        gc += __shfl_xor(gc, o);
    }

    if (lane == 0) {
        const int w = (2 * (int)counts[c] == cnt) ? 1 : 0;  // manifold test (exact ints)
        rows[row]             = w ? (sp_sum - xg_sum) : 0.f; // weighted BCE row sum
        rows[NROWS + row]     = (float)w;
        rows[2 * NROWS + row] = pp;
        rows[3 * NROWS + row] = nn;
        rows[4 * NROWS + row] = (float)gc;
    }
}

// Deterministic final reduction: 5 waves, each sums one 16384-float array on the
// matrix pipe: D = A(16x4 f32) x ones(4x16) + C accumulates row-sums in f32.
__global__ __launch_bounds__(256)
void final_reduce(const float* __restrict__ rows, float* __restrict__ out) {
    const int wave = threadIdx.x >> 5;
    const int lane = threadIdx.x & 31;
    __shared__ float sums[8];

    if (wave < 5) {
        const float* src = rows + wave * NROWS;
        v8f acc = {};
        v2f bones; bones.x = 1.f; bones.y = 1.f;
        for (int t = 0; t < NROWS / 64; ++t) {
            const int base = t * 64;
            v2f afrag;
            afrag.x = src[base + lane];
            afrag.y = src[base + 32 + lane];
            acc = __builtin_amdgcn_wmma_f32_16x16x4_f32(
                false, afrag, false, bones, (short)0, acc, false, false);
        }
        // per-lane: sum of 8 accumulator rows of this lane's column; columns are
        // identical (B all-ones), column total = lanes L and L^16 combined.
        float s = acc[0] + acc[1] + acc[2] + acc[3] + acc[4] + acc[5] + acc[6] + acc[7];
        float tot = s + __shfl_xor(s, 16);
        if (lane == 0) sums[wave] = tot;
    }
    __syncthreads();
    if (threadIdx.x == 0) {
        const float A  = sums[0];   // sum_i w_i * bce_row_i
        const float W  = sums[1];   // sum_i w_i
        const float Pp = sums[2];   // sum sp(-pos)*gm
        const float Nn = sums[3];   // sum sp(neg)*gm
        const float G  = sums[4];   // sum gm
        out[0] = A / fmaxf(W * (float)MPAIR, 1.f);
        const float cg = fmaxf(G, 1.f);
        out[1] = Pp / cg;
        out[2] = Nn / cg;
    }
}

// ---------------- host launcher ----------------

extern "C" void kernel_launch(void* const* d_in, const int* in_sizes, int n_in,
                              void* d_out, int out_size, void* d_ws, size_t ws_size,
                              hipStream_t stream) {
    const float*     pred = (const float*)d_in[0];     // (16384, 256) f32
    // d_in[1] = points (16384,3) f32 — only its shape matters (num_pts); unused.
    const long long* knn  = (const long long*)d_in[2]; // (16384, 17) i64
    const long long* tri  = (const long long*)d_in[3]; // (32768, 3) i64

    uint8_t* ws = (uint8_t*)d_ws;
    unsigned long long* keys   = (unsigned long long*)(ws + OFF_KEYS);
    unsigned*           counts = (unsigned*)(ws + OFF_COUNTS);
    unsigned*           bits   = (unsigned*)(ws + OFF_BITS);
    float*              rows   = (float*)(ws + OFF_ROWS);
    float*              out    = (float*)d_out;

    // 1) zero counts + edge bitmask (contiguous region)
    zero_words<<<(ZERO_WORDS + 255) / 256, 256, 0, stream>>>(counts, ZERO_WORDS);
    // 2) triangle keys (unsorted)
    tri_keys<<<NTRI / 256, 256, 0, stream>>>(tri, keys);
    // 3) edge adjacency bitmask (32 MB, L2-resident)
    edge_bits<<<NTRI / 256, 256, 0, stream>>>(tri, bits);
    // 4) bitonic sort of 32768 u64 keys (120 L2-resident passes)
    for (int k = 2; k <= NTRI; k <<= 1)
        for (int j = k >> 1; j >= 1; j >>= 1)
            bitonic_step<<<NTRI / 256, 256, 0, stream>>>(keys, j, k);
    // 5) dedup + per-vertex triangle counts
    count_unique<<<NTRI / 256, 256, 0, stream>>>(keys, counts);
    // 6) main per-row pass (one wave32 per row; streams pred_logits once)
    main_rows<<<NROWS / 8, 256, 0, stream>>>(pred, knn, keys, counts, bits, rows);
    // 7) deterministic WMMA reduction -> 3 outputs
    final_reduce<<<1, 256, 0, stream>>>(rows, out);
}